// GraphEncoder_64673617543327
// MI455X (gfx1250) — compile-verified
//
#include <hip/hip_runtime.h>
#include <hip/hip_bf16.h>

typedef float v2f __attribute__((ext_vector_type(2)));
typedef float v8f __attribute__((ext_vector_type(8)));

#define N_NODES 50000
#define N_GRAPHS 256
#define DIM 128
#define N_EDGES 800000

// ---------------------------------------------------------------------------
// Zero-fill (graph-capture-safe replacement for hipMemsetAsync)
// ---------------------------------------------------------------------------
__global__ void fill_zero_kernel(float* __restrict__ p, int n) {
    int i = blockIdx.x * blockDim.x + threadIdx.x;
    if (i < n) p[i] = 0.0f;
}

// ---------------------------------------------------------------------------
// Edge scatter: agg[dst] += h[src]; deg[dst] += 1
// One wave per edge; each lane moves a float4 chunk (32 lanes * 4 = 128).
// ---------------------------------------------------------------------------
__global__ void sage_scatter_kernel(const float* __restrict__ h,
                                    const int* __restrict__ src,
                                    const int* __restrict__ dst,
                                    float* __restrict__ agg,
                                    float* __restrict__ deg,
                                    int n_edges) {
    int tid = blockIdx.x * blockDim.x + threadIdx.x;
    int e = tid >> 5;
    if (e >= n_edges) return;
    int lane = tid & 31;
    int c = lane * 4;
    int s = src[e];
    int d = dst[e];
    const float4 v = *(const float4*)(h + (size_t)s * DIM + c);
    float* ap = agg + (size_t)d * DIM + c;
    atomicAdd(ap + 0, v.x);
    atomicAdd(ap + 1, v.y);
    atomicAdd(ap + 2, v.z);
    atomicAdd(ap + 3, v.w);
    if (lane == 0) atomicAdd(deg + d, 1.0f);
}

// ---------------------------------------------------------------------------
// Fused SAGE GEMM:  out = relu( (agg/max(deg,1)) @ Wl + bl + h @ Wr )
// fp32 WMMA 16x16x4. One wave computes a 16-row strip x all 128 columns.
// A layout (16x4 f32): lane L holds rows m = L&15; VGPR pair = K {2*(L>>4), +1}
// B layout (4x16):     lane L holds col n = L&15; VGPR pair = K {2*(L>>4), +1}
// C/D (16x16):         VGPR r: lanes 0-15 -> M=r, lanes 16-31 -> M=r+8; N=L&15
// ---------------------------------------------------------------------------
__global__ void sage_gemm_kernel(const float* __restrict__ agg,
                                 const float* __restrict__ deg,
                                 const float* __restrict__ h,
                                 const float* __restrict__ Wl,
                                 const float* __restrict__ bl,
                                 const float* __restrict__ Wr,
                                 float* __restrict__ out,
                                 int n_nodes) {
    const int lane = threadIdx.x & 31;
    const int waveInBlock = threadIdx.x >> 5;
    const int wavesPerBlock = blockDim.x >> 5;
    const int mtile = blockIdx.x * wavesPerBlock + waveInBlock;
    const int m0 = mtile * 16;
    if (m0 >= n_nodes) return;          // wave-uniform: EXEC stays all-ones

    const int mrow = lane & 15;         // A/C row within tile, B/C column
    const int hi = lane >> 4;           // K-pair selector for A/B
    const int m = m0 + mrow;

    const float rdeg = 1.0f / fmaxf(deg[m], 1.0f);

    v8f acc[8];
#pragma unroll
    for (int t = 0; t < 8; ++t) acc[t] = (v8f)(0.0f);

    for (int kk = 0; kk < DIM; kk += 4) {
        const int ka = kk + 2 * hi;     // this lane's first K of the pair
        v2f a1 = *(const v2f*)(agg + (size_t)m * DIM + ka);
        a1.x *= rdeg;
        a1.y *= rdeg;
        v2f a2 = *(const v2f*)(h + (size_t)m * DIM + ka);
#pragma unroll
        for (int t = 0; t < 8; ++t) {
            const int n = t * 16 + mrow;
            v2f b1, b2;
            b1.x = Wl[(size_t)ka * DIM + n];
            b1.y = Wl[(size_t)(ka + 1) * DIM + n];
            b2.x = Wr[(size_t)ka * DIM + n];
            b2.y = Wr[(size_t)(ka + 1) * DIM + n];
            acc[t] = __builtin_amdgcn_wmma_f32_16x16x4_f32(
                false, a1, false, b1, (short)0, acc[t], false, false);
            acc[t] = __builtin_amdgcn_wmma_f32_16x16x4_f32(
                false, a2, false, b2, (short)0, acc[t], false, false);
        }
    }

    // Epilogue: bias + ReLU + store
#pragma unroll
    for (int t = 0; t < 8; ++t) {
        const int col = t * 16 + mrow;
        const float b = bl[col];
#pragma unroll
        for (int r = 0; r < 8; ++r) {
            const int row = m0 + r + 8 * hi;
            float v = acc[t][r] + b;
            out[(size_t)row * DIM + col] = fmaxf(v, 0.0f);
        }
    }
}

// ---------------------------------------------------------------------------
// Readout scatter: per (node, feature) -> atomic mean-sum and atomic max.
// h >= 0 after ReLU, so float bit pattern is monotonic as unsigned:
// atomicMax on the uint view == float max, and 0-init matches the
// reference's "empty graph -> 0" semantics.
// ---------------------------------------------------------------------------
__global__ void readout_scatter_kernel(const float* __restrict__ h,
                                       const int* __restrict__ batch,
                                       float* __restrict__ g_sum,
                                       float* __restrict__ g_cnt,
                                       unsigned int* __restrict__ g_max,
                                       int n_nodes) {
    int tid = blockIdx.x * blockDim.x + threadIdx.x;
    int n = tid >> 7;                   // DIM = 128 features per node
    if (n >= n_nodes) return;
    int f = tid & 127;
    int g = batch[n];
    float v = h[(size_t)n * DIM + f];
    atomicAdd(&g_sum[g * DIM + f], v);
    atomicMax(&g_max[g * DIM + f], __float_as_uint(v));
    if (f == 0) atomicAdd(&g_cnt[g], 1.0f);
}

__global__ void readout_finalize_kernel(const float* __restrict__ g_sum,
                                        const float* __restrict__ g_cnt,
                                        const unsigned int* __restrict__ g_max,
                                        float* __restrict__ out) {
    int tid = blockIdx.x * blockDim.x + threadIdx.x;
    if (tid >= N_GRAPHS * DIM) return;
    int g = tid >> 7;
    int f = tid & 127;
    float c = fmaxf(g_cnt[g], 1.0f);
    out[(size_t)g * (2 * DIM) + f] = g_sum[g * DIM + f] / c;
    out[(size_t)g * (2 * DIM) + DIM + f] = __uint_as_float(g_max[g * DIM + f]);
}

// ---------------------------------------------------------------------------
// Launcher
// ---------------------------------------------------------------------------
extern "C" void kernel_launch(void* const* d_in, const int* in_sizes, int n_in,
                              void* d_out, int out_size, void* d_ws, size_t ws_size,
                              hipStream_t stream) {
    (void)in_sizes; (void)n_in; (void)out_size; (void)ws_size;

    const float* x     = (const float*)d_in[0];
    const int*   ei    = (const int*)d_in[1];
    const int*   batch = (const int*)d_in[2];
    const float* Wl[3] = {(const float*)d_in[3], (const float*)d_in[6], (const float*)d_in[9]};
    const float* bl[3] = {(const float*)d_in[4], (const float*)d_in[7], (const float*)d_in[10]};
    const float* Wr[3] = {(const float*)d_in[5], (const float*)d_in[8], (const float*)d_in[11]};

    const int* srcI = ei;             // edge_index[0, :]
    const int* dstI = ei + N_EDGES;   // edge_index[1, :]

    // Workspace carve-up (floats): h1 | h2 | agg | deg | g_sum | g_cnt | g_max
    float* ws    = (float*)d_ws;
    float* h1    = ws;
    float* h2    = h1 + (size_t)N_NODES * DIM;
    float* agg   = h2 + (size_t)N_NODES * DIM;
    float* deg   = agg + (size_t)N_NODES * DIM;   // contiguous after agg
    float* g_sum = deg + N_NODES;
    float* g_cnt = g_sum + (size_t)N_GRAPHS * DIM;
    unsigned int* g_max = (unsigned int*)(g_cnt + N_GRAPHS);

    const int aggDegCount = N_NODES * DIM + N_NODES;
    const int readoutCount = N_GRAPHS * DIM + N_GRAPHS + N_GRAPHS * DIM;

    const float* hin = x;
    float* houts[3] = {h1, h2, h1};

    const int wavesPerBlock = 4;
    const int mtiles = N_NODES / 16;  // 50000 / 16 = 3125 exactly, no tail
    dim3 gemmGrid((mtiles + wavesPerBlock - 1) / wavesPerBlock);
    dim3 gemmBlock(32 * wavesPerBlock);

    for (int l = 0; l < 3; ++l) {
        fill_zero_kernel<<<(aggDegCount + 255) / 256, 256, 0, stream>>>(agg, aggDegCount);
        long long scatterThreads = (long long)N_EDGES * 32;
        sage_scatter_kernel<<<(unsigned)((scatterThreads + 255) / 256), 256, 0, stream>>>(
            hin, srcI, dstI, agg, deg, N_EDGES);
        sage_gemm_kernel<<<gemmGrid, gemmBlock, 0, stream>>>(
            agg, deg, hin, Wl[l], bl[l], Wr[l], houts[l], N_NODES);
        hin = houts[l];
    }

    fill_zero_kernel<<<(readoutCount + 255) / 256, 256, 0, stream>>>(g_sum, readoutCount);
    readout_scatter_kernel<<<(N_NODES * DIM + 255) / 256, 256, 0, stream>>>(
        hin, batch, g_sum, g_cnt, g_max, N_NODES);
    readout_finalize_kernel<<<(N_GRAPHS * DIM + 255) / 256, 256, 0, stream>>>(
        g_sum, g_cnt, g_max, (float*)d_out);
}